// FullNet_35708358099468
// MI455X (gfx1250) — compile-verified
//
#include <hip/hip_runtime.h>
#include <math.h>

typedef float v2f __attribute__((ext_vector_type(2)));
typedef float v8f __attribute__((ext_vector_type(8)));

#define VDDc    1.8f
#define VRc     0.225f      // VDD / 2^3
#define VREFc   0.225f
#define AMPLc   10000.0f
#define DELTAc  1e-30f

// d_ws float layout
#define WS_STATS 0    // 48: sums[24], sumsq[24]
#define WS_WN    48   // 6
#define WS_SS    64   // scale[24] @64, shift[24] @88
#define WS_A     128  // A[6][32]

// ---------------------------------------------------------------- helpers
__device__ __forceinline__ void compute_y(const float* __restrict__ xrow,
                                          const float* __restrict__ Ag,
                                          float* y) {
  // y[k*6+i] = sum_j x[k*32+j] * A[i][j]; A reads are wave-uniform -> s_load
#pragma unroll
  for (int k = 0; k < 4; ++k) {
    const float4* xp = reinterpret_cast<const float4*>(xrow + k * 32);
    float4 xv[8];
#pragma unroll
    for (int q = 0; q < 8; ++q) xv[q] = xp[q];
#pragma unroll
    for (int i = 0; i < 6; ++i) {
      float acc = 0.0f;
#pragma unroll
      for (int q = 0; q < 8; ++q) {
        const float* Ar = Ag + i * 32 + q * 4;
        acc = fmaf(xv[q].x, Ar[0], acc);
        acc = fmaf(xv[q].y, Ar[1], acc);
        acc = fmaf(xv[q].z, Ar[2], acc);
        acc = fmaf(xv[q].w, Ar[3], acc);
      }
      y[k * 6 + i] = acc;
    }
  }
}

// ---------------------------------------------------------------- pass 0
__global__ void init_kernel(const float* __restrict__ pvals,
                            const float* __restrict__ Wt,
                            const float* __restrict__ wnoise,
                            float* __restrict__ ws,
                            float* __restrict__ out_wn) {
  const int t = threadIdx.x;
  if (t < 48) ws[WS_STATS + t] = 0.0f;
  if (t < 6) {
    float v = Wt[t] + wnoise[t];
    ws[WS_WN + t] = v;
    out_wn[t] = v;                     // Wn output (6 floats)
  }
  if (t < 192) {                       // A[6][32]
    const int i = t >> 5, j = t & 31;
    const int m = j & 15;
    const bool upper = (j >= 16);
    const float p = pvals[(i < 3) ? i : (i - 3)];
    const float th = p * (float)m * (6.28318530717958647692f / 16.0f);
    float s, c;
    sincosf(th, &s, &c);
    float v;
    if (i < 3) v = upper ? s : c;      // [Ar | -Ai_neg=sin]
    else       v = upper ? c : -s;     // [Ai=-sin | Ar]
    ws[WS_A + i * 32 + j] = v;
  }
}

// ---------------------------------------------------------------- pass 1
__global__ __launch_bounds__(256) void stats_kernel(const float* __restrict__ x,
                                                    float* ws) {
  __shared__ float sAcc[48];
  const int t = threadIdx.x;
  if (t < 48) sAcc[t] = 0.0f;
  __syncthreads();
  const size_t row = (size_t)blockIdx.x * 256 + t;
  float y[24];
  compute_y(x + row * 128, ws + WS_A, y);
#pragma unroll
  for (int c = 0; c < 24; ++c) {
    atomicAdd(&sAcc[c], y[c]);
    atomicAdd(&sAcc[24 + c], y[c] * y[c]);
  }
  __syncthreads();
  if (t < 48) atomicAdd(&ws[WS_STATS + t], sAcc[t]);
}

// ---------------------------------------------------------------- pass 2
__global__ void finalize_kernel(const float* __restrict__ gamma,
                                const float* __restrict__ beta,
                                float* __restrict__ ws, int B) {
  const int c = threadIdx.x;
  if (c < 24) {
    const float invB = 1.0f / (float)B;
    const float mean = ws[WS_STATS + c] * invB;
    const float var  = ws[WS_STATS + 24 + c] * invB - mean * mean;
    const float inv  = rsqrtf(var + 1e-5f);
    const float sc   = gamma[c] * inv;
    ws[WS_SS + c]      = sc;
    ws[WS_SS + 24 + c] = beta[c] - mean * sc;
  }
}

// ---------------------------------------------------------------- pass 3
#define QS 25   // padded stride of q-stage (24 channels)
#define HS 65   // padded stride of h-stage (64 channels)

__global__ __launch_bounds__(256) void main_kernel(
    const float* __restrict__ x, const float* ws,
    const float* __restrict__ l1w, const float* __restrict__ l1b,
    const float* __restrict__ l2w, const float* __restrict__ l2b,
    float* __restrict__ out, float* __restrict__ Qout, float* __restrict__ VinOut) {
  __shared__ float sW1T[24 * 64];   // W1T[k][n] = l1_w[n][k]
  __shared__ float sW2T[64 * 32];   // W2T[k][n] = l2_w[n][k]
  __shared__ float sB1[64];
  __shared__ float sB2[32];
  __shared__ float sQ[8 * 32 * QS]; // per-wave 32-row q tiles
  __shared__ float sH[8 * 32 * HS]; // per-wave 32-row h tiles

  const int tid = threadIdx.x;
  for (int idx = tid; idx < 24 * 64; idx += 256) {
    const int k = idx >> 6, n = idx & 63;
    sW1T[k * 64 + n] = l1w[n * 24 + k];
  }
  for (int idx = tid; idx < 64 * 32; idx += 256) {
    const int k = idx >> 5, n = idx & 31;
    sW2T[k * 32 + n] = l2w[n * 64 + k];
  }
  if (tid < 64) sB1[tid] = l1b[tid];
  if (tid < 32) sB2[tid] = l2b[tid];

  const int wave = tid >> 5;
  const int lane = tid & 31;
  const size_t row = (size_t)blockIdx.x * 256 + tid;

  // ---- elementwise front: y -> BN -> Vin -> SAR bits -> q (one row/lane) ----
  float y[24];
  compute_y(x + row * 128, ws + WS_A, y);

  const float w0 = ws[WS_WN + 0], w1 = ws[WS_WN + 1], w2 = ws[WS_WN + 2];
  const float w3 = ws[WS_WN + 3], w4 = ws[WS_WN + 4], w5 = ws[WS_WN + 5];

#pragma unroll
  for (int c = 0; c < 24; ++c) {
    const float sc = ws[WS_SS + c], sh = ws[WS_SS + 24 + c];
    const float vin = (fmaf(y[c], sc, sh) + 1.0f) * (VDDc * 0.5f);
    VinOut[row * 24 + c] = vin;
    const float q2 = tanhf(AMPLc * (vin - w5 * VREFc + DELTAc));
    const float q1 = tanhf(AMPLc * (vin - (w4 * VREFc + 0.5f * (q2 + 1.0f) * w3 * VRc) + DELTAc));
    const float q0 = tanhf(AMPLc * (vin - (w2 * VREFc + 0.5f * (q1 + 1.0f) * w1 * VRc
                                                       + 0.5f * (q2 + 1.0f) * w0 * VRc) + DELTAc));
    const size_t qb = (row * 24 + c) * 3;
    Qout[qb + 0] = q0; Qout[qb + 1] = q1; Qout[qb + 2] = q2;
    const float qv = 0.5f * (q0 + 1.0f) * VRc
                   + 0.5f * (q1 + 1.0f) * (2.0f * VRc)
                   + 0.5f * (q2 + 1.0f) * (4.0f * VRc);
    sQ[(size_t)tid * QS + c] = qv;
  }
  __syncthreads();  // covers weight staging + q staging

  // ---- layer 1: (32x24) @ (24x64) via v_wmma_f32_16x16x4_f32 ----
  const int lh = lane >> 4;          // lane half selects K pair / M offset
  const int lm = lane & 15;
  const v8f vzero = {};
  v8f acc1[2][4];
#pragma unroll
  for (int t = 0; t < 2; ++t)
#pragma unroll
    for (int nt = 0; nt < 4; ++nt) acc1[t][nt] = vzero;

#pragma unroll
  for (int ks = 0; ks < 6; ++ks) {
    const int kb = ks * 4 + lh * 2;
    v2f a[2];
#pragma unroll
    for (int t = 0; t < 2; ++t) {
      const float* qrow = &sQ[(size_t)(wave * 32 + t * 16 + lm) * QS];
      a[t].x = qrow[kb]; a[t].y = qrow[kb + 1];
    }
#pragma unroll
    for (int nt = 0; nt < 4; ++nt) {
      const int n = nt * 16 + lm;
      v2f b; b.x = sW1T[kb * 64 + n]; b.y = sW1T[(kb + 1) * 64 + n];
#pragma unroll
      for (int t = 0; t < 2; ++t)
        acc1[t][nt] = __builtin_amdgcn_wmma_f32_16x16x4_f32(
            false, a[t], false, b, (short)0, acc1[t][nt], false, false);
    }
  }

  // bias + relu, re-stage h in row-major for layer-2 A operand
#pragma unroll
  for (int t = 0; t < 2; ++t)
#pragma unroll
    for (int nt = 0; nt < 4; ++nt) {
      const int n = nt * 16 + lm;
      const float bias = sB1[n];
#pragma unroll
      for (int v = 0; v < 8; ++v) {
        const int m = lh * 8 + v;
        sH[(size_t)(wave * 32 + t * 16 + m) * HS + n] =
            fmaxf(acc1[t][nt][v] + bias, 0.0f);
      }
    }
  __syncthreads();

  // ---- layer 2: (32x64) @ (64x32) via WMMA ----
  v8f acc2[2][2];
#pragma unroll
  for (int t = 0; t < 2; ++t)
#pragma unroll
    for (int nt = 0; nt < 2; ++nt) acc2[t][nt] = vzero;

#pragma unroll
  for (int ks = 0; ks < 16; ++ks) {
    const int kb = ks * 4 + lh * 2;
    v2f a[2];
#pragma unroll
    for (int t = 0; t < 2; ++t) {
      const float* hrow = &sH[(size_t)(wave * 32 + t * 16 + lm) * HS];
      a[t].x = hrow[kb]; a[t].y = hrow[kb + 1];
    }
#pragma unroll
    for (int nt = 0; nt < 2; ++nt) {
      const int n = nt * 16 + lm;
      v2f b; b.x = sW2T[kb * 32 + n]; b.y = sW2T[(kb + 1) * 32 + n];
#pragma unroll
      for (int t = 0; t < 2; ++t)
        acc2[t][nt] = __builtin_amdgcn_wmma_f32_16x16x4_f32(
            false, a[t], false, b, (short)0, acc2[t][nt], false, false);
    }
  }

  const size_t rowBase = (size_t)blockIdx.x * 256 + wave * 32;
#pragma unroll
  for (int t = 0; t < 2; ++t)
#pragma unroll
    for (int nt = 0; nt < 2; ++nt) {
      const int n = nt * 16 + lm;
      const float bias = sB2[n];
#pragma unroll
      for (int v = 0; v < 8; ++v) {
        const int m = lh * 8 + v;
        out[(rowBase + t * 16 + m) * 32 + n] = acc2[t][nt][v] + bias;
      }
    }
}

// ---------------------------------------------------------------- launch
extern "C" void kernel_launch(void* const* d_in, const int* in_sizes, int n_in,
                              void* d_out, int out_size, void* d_ws, size_t ws_size,
                              hipStream_t stream) {
  const float* x    = (const float*)d_in[0];
  const float* pv   = (const float*)d_in[1];
  const float* Wt   = (const float*)d_in[2];
  const float* wnz  = (const float*)d_in[3];
  const float* gam  = (const float*)d_in[4];
  const float* bet  = (const float*)d_in[5];
  const float* l1w  = (const float*)d_in[6];
  const float* l1b  = (const float*)d_in[7];
  const float* l2w  = (const float*)d_in[8];
  const float* l2b  = (const float*)d_in[9];
  float* ws  = (float*)d_ws;
  float* out = (float*)d_out;

  const int B = in_sizes[0] / 128;            // 262144
  float* Qout  = out + (size_t)B * 32;        // (B,24,3)
  float* Vin   = out + (size_t)B * 104;       // (B,24)
  float* WnOut = out + (size_t)B * 128;       // (6,1)
  const int nblk = B / 256;

  hipLaunchKernelGGL(init_kernel,     dim3(1),    dim3(256), 0, stream, pv, Wt, wnz, ws, WnOut);
  hipLaunchKernelGGL(stats_kernel,    dim3(nblk), dim3(256), 0, stream, x, ws);
  hipLaunchKernelGGL(finalize_kernel, dim3(1),    dim3(32),  0, stream, gam, bet, ws, B);
  hipLaunchKernelGGL(main_kernel,     dim3(nblk), dim3(256), 0, stream,
                     x, ws, l1w, l1b, l2w, l2b, out, Qout, Vin);
}